// BetterBot_44169443672375
// MI455X (gfx1250) — compile-verified
//
#include <hip/hip_runtime.h>

// ---------------------------------------------------------------------------
// MI455X (gfx1250) implementation: one wave32 per batch item, all GEMMs as
// V_WMMA_F32_16X16X4_F32 (tokens padded 15->16 as M), layout shuffles via
// per-wave LDS tiles (stride 20 floats: conflict-free + 16B aligned rows),
// in-lane softmax, per-wave weight caching in B-layout registers.
// ---------------------------------------------------------------------------

#define NB    65536
#define DM    8
#define AOUT  20
#define NL    5
#define RS    20                 // LDS row stride (floats)
#define WPB   8                  // waves per block
#define BLOCK (WPB * 32)

typedef float v2f __attribute__((ext_vector_type(2)));
typedef float v4f __attribute__((ext_vector_type(4)));
typedef float v8f __attribute__((ext_vector_type(8)));

static __device__ __forceinline__ v8f wmma4(v2f a, v2f b, v8f c) {
  // D(16x16,f32) = A(16x4,f32) * B(4x16,f32) + C
  return __builtin_amdgcn_wmma_f32_16x16x4_f32(false, a, false, b,
                                               (short)0, c, false, false);
}
static __device__ __forceinline__ void lfence() {
  // DS ops are in-order within a wave on CDNA5; we only need to stop the
  // compiler from reordering LDS traffic across phase boundaries.
  asm volatile("" ::: "memory");
}
static __device__ __forceinline__ v2f ld2(const float* p) { return *(const v2f*)p; }
static __device__ __forceinline__ void st2(float* p, v2f v) { *(v2f*)p = v; }
static __device__ __forceinline__ v4f ld4(const float* p) { return *(const v4f*)p; }
static __device__ __forceinline__ void st4(float* p, v4f v) { *(v4f*)p = v; }
static __device__ __forceinline__ v8f splat8(float x) {
  v8f v;
#pragma unroll
  for (int i = 0; i < 8; ++i) v[i] = x;
  return v;
}

__global__ __launch_bounds__(BLOCK) void
betterbot_wmma_kernel(const int* __restrict__ dice_type,
                      const int* __restrict__ dice_star,
                      const int* __restrict__ summon_lvl,
                      const float* __restrict__ emb_dice,
                      const float* __restrict__ emb_star,
                      const float* __restrict__ emb_btns,
                      const float* __restrict__ Wout, const float* __restrict__ bout,
                      const float* __restrict__ Wqkv0, const float* __restrict__ bqkv0,
                      const float* __restrict__ Wo0,   const float* __restrict__ bo0,
                      const float* __restrict__ Wl0,   const float* __restrict__ bl0,
                      const float* __restrict__ Wqkv1, const float* __restrict__ bqkv1,
                      const float* __restrict__ Wo1,   const float* __restrict__ bo1,
                      const float* __restrict__ Wl1,   const float* __restrict__ bl1,
                      float* __restrict__ out)
{
  __shared__ float lds[WPB][4 * 16 * RS];   // 4 tiles per wave: x, qk, v, mat

  const int lane = threadIdx.x & 31;
  const int wv   = threadIdx.x >> 5;
  const int n    = lane & 15;    // col / token index within half-wave
  const int hl   = lane >> 4;    // lane half (WMMA K/M split)

  float* xb  = &lds[wv][0 * 16 * RS];
  float* qkb = &lds[wv][1 * 16 * RS];
  float* vb  = &lds[wv][2 * 16 * RS];
  float* mb  = &lds[wv][3 * 16 * RS];

  // ---- per-wave weight setup (amortized over the item loop) -------------
  const float* Wq[2] = {Wqkv0, Wqkv1};
  const float* bq[2] = {bqkv0, bqkv1};
  const float* Wo[2] = {Wo0, Wo1};
  const float* bo[2] = {bo0, bo1};
  const float* Wl[2] = {Wl0, Wl1};
  const float* bl[2] = {bl0, bl1};

  // B-layout (4x16): lane half splits K range; reg r holds K = 4s + 2*hl + r.
  v2f Bq_[2][2][2];   float bqC[2][2];
  v2f Bo_[2][2];      float boC[2];
  v2f Bl_[2][2];      float blC[2];
#pragma unroll
  for (int L = 0; L < 2; ++L) {
#pragma unroll
    for (int t = 0; t < 2; ++t)
#pragma unroll
      for (int s = 0; s < 2; ++s) {
        int col = 16 * t + n;                 // output row of Wqkv (N index)
        int k0  = 4 * s + 2 * hl;             // input dim (K index)
        bool ok = col < 3 * DM;
        v2f w = ld2(Wq[L] + (ok ? col : 0) * DM + k0);
        if (!ok) { w.x = 0.f; w.y = 0.f; }
        Bq_[L][t][s] = w;
      }
    bqC[L][0] = bq[L][n];
    bqC[L][1] = (n < 8) ? bq[L][16 + n] : 0.f;
#pragma unroll
    for (int s = 0; s < 2; ++s) {
      int k0 = 4 * s + 2 * hl;
      bool ok = n < 8;
      v2f wo = ld2(Wo[L] + (ok ? n : 0) * DM + k0);
      v2f wl = ld2(Wl[L] + (ok ? n : 0) * DM + k0);
      if (!ok) { wo.x = 0.f; wo.y = 0.f; wl.x = 0.f; wl.y = 0.f; }
      Bo_[L][s] = wo;
      Bl_[L][s] = wl;
    }
    boC[L] = (n < 8) ? bo[L][n] : 0.f;
    blC[L] = (n < 8) ? bl[L][n] : 0.f;
  }
  // Output projection row for this lane (lanes 0..19 -> output features).
  float wr[8];
  float bout_l;
  {
    bool ok = lane < AOUT;
    const float* p = Wout + (ok ? lane : 0) * DM;
#pragma unroll
    for (int d = 0; d < 8; ++d) wr[d] = ok ? p[d] : 0.f;
    bout_l = ok ? bout[lane] : 0.f;
  }
  // Token-fixed gather sources (token = n; token 15 is padding -> dummy row).
  const int*   iptr;
  int          ioff;
  const float* ebase;
  if (n < 5)       { iptr = dice_type;  ioff = n;      ebase = emb_dice; }
  else if (n < 10) { iptr = dice_star;  ioff = n - 5;  ebase = emb_star; }
  else if (n < 15) { iptr = summon_lvl; ioff = n - 10; ebase = emb_btns; }
  else             { iptr = dice_type;  ioff = 0;      ebase = emb_dice; }

  const int gw = blockIdx.x * WPB + wv;
  const int nw = gridDim.x * WPB;

  for (int b = gw; b < NB; b += nw) {
    // ---- gather x (A-layout regs come straight from LDS) ----------------
    int idx = iptr[b * NL + ioff];
    const float* row = ebase + idx * DM;
    st2(xb + n * RS + 2 * hl,     ld2(row + 2 * hl));
    st2(xb + n * RS + 4 + 2 * hl, ld2(row + 4 + 2 * hl));
    lfence();

    v8f xC;  // layer output in C layout (live across layer loop tail)
#pragma unroll
    for (int L = 0; L < 2; ++L) {
      // x in A layout: lane = token, regs = K slice
      v2f xA0 = ld2(xb + n * RS + 0 + 2 * hl);
      v2f xA1 = ld2(xb + n * RS + 4 + 2 * hl);

      // qkv projection: tile0 = q|k cols 0..15, tile1 = v cols 0..7
      v8f qkC = splat8(bqC[L][0]);
      qkC = wmma4(xA0, Bq_[L][0][0], qkC);
      qkC = wmma4(xA1, Bq_[L][0][1], qkC);
      v8f vC = splat8(bqC[L][1]);
      vC = wmma4(xA0, Bq_[L][1][0], vC);
      vC = wmma4(xA1, Bq_[L][1][1], vC);
#pragma unroll
      for (int r = 0; r < 8; ++r) {
        qkb[(r + 8 * hl) * RS + n] = qkC[r];
        vb [(r + 8 * hl) * RS + n] = vC[r];
      }
      lfence();

      // ---- attention: per head, scores (1 WMMA) + softmax + o (4 WMMA) --
      v8f oC = splat8(0.f);
#pragma unroll
      for (int h = 0; h < 2; ++h) {
        v2f qA = ld2(qkb + n * RS + 4 * h + 2 * hl);        // A: q_h
        v2f kB = ld2(qkb + n * RS + 8 + 4 * h + 2 * hl);    // B: k_h^T
        v8f sC = splat8(0.f);
        sC = wmma4(qA, kB, sC);
#pragma unroll
        for (int r = 0; r < 8; ++r)
          mb[(r + 8 * hl) * RS + n] = sC[r] * 0.5f;         // * hd^-0.5
        lfence();

        // in-lane softmax over row q = n (k = 0..14; col 15 masked to 0)
        v4f a0 = ld4(mb + n * RS + 0);
        v4f a1 = ld4(mb + n * RS + 4);
        v4f a2 = ld4(mb + n * RS + 8);
        v4f a3 = ld4(mb + n * RS + 12);
        float rv[16];
        rv[0]=a0.x; rv[1]=a0.y; rv[2]=a0.z; rv[3]=a0.w;
        rv[4]=a1.x; rv[5]=a1.y; rv[6]=a1.z; rv[7]=a1.w;
        rv[8]=a2.x; rv[9]=a2.y; rv[10]=a2.z; rv[11]=a2.w;
        rv[12]=a3.x; rv[13]=a3.y; rv[14]=a3.z; rv[15]=0.f;
        float mx = rv[0];
#pragma unroll
        for (int i = 1; i < 15; ++i) mx = fmaxf(mx, rv[i]);
        float sum = 0.f;
#pragma unroll
        for (int i = 0; i < 15; ++i) { float e = __expf(rv[i] - mx); rv[i] = e; sum += e; }
        float inv = 1.f / sum;
#pragma unroll
        for (int i = 0; i < 15; ++i) rv[i] *= inv;
        v4f o0, o1, o2, o3;
        o0.x=rv[0]; o0.y=rv[1]; o0.z=rv[2]; o0.w=rv[3];
        o1.x=rv[4]; o1.y=rv[5]; o1.z=rv[6]; o1.w=rv[7];
        o2.x=rv[8]; o2.y=rv[9]; o2.z=rv[10]; o2.w=rv[11];
        o3.x=rv[12]; o3.y=rv[13]; o3.z=rv[14]; o3.w=0.f;
        st4(mb + n * RS + 0,  o0);
        st4(mb + n * RS + 4,  o1);
        st4(mb + n * RS + 8,  o2);
        st4(mb + n * RS + 12, o3);
        lfence();

        // o += attn_h @ v_h, K=16 chained; v placed in cols 4h..4h+3
        bool inh = (n >= 4 * h) && (n < 4 * h + 4);
#pragma unroll
        for (int j = 0; j < 4; ++j) {
          v2f aA = ld2(mb + n * RS + 4 * j + 2 * hl);
          int k0 = 4 * j + 2 * hl;
          float b0 = vb[(k0 + 0) * RS + n];
          float b1 = vb[(k0 + 1) * RS + n];
          v2f vB;
          vB.x = inh ? b0 : 0.f;
          vB.y = inh ? b1 : 0.f;
          oC = wmma4(aA, vB, oC);
        }
        lfence();
      }

      // ---- Wo + residual:  x1 = x + o@Wo^T + bo -------------------------
#pragma unroll
      for (int r = 0; r < 8; ++r) qkb[(r + 8 * hl) * RS + n] = oC[r];
      lfence();
      v2f oA0 = ld2(qkb + n * RS + 0 + 2 * hl);
      v2f oA1 = ld2(qkb + n * RS + 4 + 2 * hl);
      v8f x1;
#pragma unroll
      for (int r = 0; r < 8; ++r) x1[r] = xb[(r + 8 * hl) * RS + n] + boC[L];
      x1 = wmma4(oA0, Bo_[L][0], x1);
      x1 = wmma4(oA1, Bo_[L][1], x1);
#pragma unroll
      for (int r = 0; r < 8; ++r) xb[(r + 8 * hl) * RS + n] = x1[r];
      lfence();

      // ---- Wl + ReLU + residual: x2 = x1 + relu(x1@Wl^T + bl) -----------
      v2f yA0 = ld2(xb + n * RS + 0 + 2 * hl);
      v2f yA1 = ld2(xb + n * RS + 4 + 2 * hl);
      v8f hC = splat8(blC[L]);
      hC = wmma4(yA0, Bl_[L][0], hC);
      hC = wmma4(yA1, Bl_[L][1], hC);
#pragma unroll
      for (int r = 0; r < 8; ++r) x1[r] += fmaxf(hC[r], 0.f);
#pragma unroll
      for (int r = 0; r < 8; ++r) xb[(r + 8 * hl) * RS + n] = x1[r];
      lfence();
      xC = x1;
    }

    // ---- mean over 15 tokens (token 15 = reg7 on upper half, excluded) --
    float s = 0.f;
#pragma unroll
    for (int r = 0; r < 8; ++r) {
      float v = xC[r];
      if (r == 7) v = (hl == 1) ? 0.f : v;
      s += v;
    }
    s += __shfl_xor(s, 16, 32);
    float xm = s * (1.0f / 15.0f);          // lane holds dim n (n<8 valid)

    if (lane < 8) mb[lane] = xm;            // stage mean vector
    lfence();
    v4f m0 = ld4(mb + 0);
    v4f m1 = ld4(mb + 4);
    float o = bout_l;
    o += m0.x * wr[0] + m0.y * wr[1] + m0.z * wr[2] + m0.w * wr[3];
    o += m1.x * wr[4] + m1.y * wr[5] + m1.z * wr[6] + m1.w * wr[7];
    if (lane < AOUT) out[b * AOUT + lane] = o;
    lfence();
  }
}

extern "C" void kernel_launch(void* const* d_in, const int* in_sizes, int n_in,
                              void* d_out, int out_size, void* d_ws, size_t ws_size,
                              hipStream_t stream) {
  (void)in_sizes; (void)n_in; (void)out_size; (void)d_ws; (void)ws_size;
  const int*   dice_type  = (const int*)d_in[0];
  const int*   dice_star  = (const int*)d_in[1];
  const int*   summon_lvl = (const int*)d_in[2];
  const float* emb_dice   = (const float*)d_in[3];
  const float* emb_star   = (const float*)d_in[4];
  const float* emb_btns   = (const float*)d_in[5];
  const float* Wout       = (const float*)d_in[6];
  const float* bout       = (const float*)d_in[7];
  const float* Wqkv0      = (const float*)d_in[8];
  const float* bqkv0      = (const float*)d_in[9];
  const float* Wo0        = (const float*)d_in[10];
  const float* bo0        = (const float*)d_in[11];
  const float* Wl0        = (const float*)d_in[12];
  const float* bl0        = (const float*)d_in[13];
  const float* Wqkv1      = (const float*)d_in[14];
  const float* bqkv1      = (const float*)d_in[15];
  const float* Wo1        = (const float*)d_in[16];
  const float* bo1        = (const float*)d_in[17];
  const float* Wl1        = (const float*)d_in[18];
  const float* bl1        = (const float*)d_in[19];

  dim3 grid(2048), block(BLOCK);  // 16384 waves, 4 items/wave
  betterbot_wmma_kernel<<<grid, block, 0, stream>>>(
      dice_type, dice_star, summon_lvl,
      emb_dice, emb_star, emb_btns, Wout, bout,
      Wqkv0, bqkv0, Wo0, bo0, Wl0, bl0,
      Wqkv1, bqkv1, Wo1, bo1, Wl1, bl1,
      (float*)d_out);
}